// CenterNet_54365696033545
// MI455X (gfx1250) — compile-verified
//
#include <hip/hip_runtime.h>
#include <hip/hip_bf16.h>

// ---------------------------------------------------------------------------
// CenterNet decode for MI455X (gfx1250, wave32).
// Stage 1: per-(batch,class) plane -> TDM load to LDS -> 5x5 maxpool NMS
//          (separable, on raw logits) -> sigmoid -> exact top-100 via LDS
//          radix select -> candidate buffer in workspace.
// Stage 2: per-batch top-100 of 8000 candidates (radix select + bitonic
//          sort of 128 keys) -> box decode at winners -> flat outputs.
// ---------------------------------------------------------------------------

#define CN_B   8
#define CN_C   80
#define CN_H   128
#define CN_W   128
#define CN_HW  (CN_H * CN_W)         // 16384
#define CN_K   100
#define CN_CAND (CN_C * CN_K)        // 8000 per batch
#define CN_STRIDE 4.0f
#define CN_INV_INPUT (1.0f / 512.0f)

typedef unsigned int       uint;
typedef unsigned long long ull;
typedef uint  u32x4 __attribute__((ext_vector_type(4)));
typedef int   i32x8 __attribute__((ext_vector_type(8)));
typedef int   i32x4 __attribute__((ext_vector_type(4)));

#if defined(__gfx1250__) && __has_builtin(__builtin_amdgcn_tensor_load_to_lds) && \
    __has_builtin(__builtin_amdgcn_s_wait_tensorcnt)
#define CN_USE_TDM 1
#else
#define CN_USE_TDM 0
#endif

__device__ __forceinline__ float cn_sigmoid(float v) {
  return 1.0f / (1.0f + __expf(-v));
}

// Exact descending radix select: returns the float-bit threshold T of the
// `need`-th largest value among s_vals[0..n), and writes back how many
// elements equal to T belong in the selection. All scores are >= 0, so the
// raw float bit pattern is monotone as a uint key.
__device__ uint cn_radix_select(const float* s_vals, int n, int need_in,
                                int* need_out, uint* s_hist, uint* s_misc,
                                int tid, int nthreads) {
  uint prefix = 0;
  int need = need_in;
  for (int pass = 3; pass >= 0; --pass) {
    const int shift = pass * 8;
    for (int i = tid; i < 256; i += nthreads) s_hist[i] = 0u;
    __syncthreads();
    const uint pmask = (pass == 3) ? 0u : (0xFFFFFFFFu << ((pass + 1) * 8));
    for (int i = tid; i < n; i += nthreads) {
      uint bts = __float_as_uint(s_vals[i]);
      if ((bts & pmask) == prefix)
        atomicAdd(&s_hist[(bts >> shift) & 255u], 1u);
    }
    __syncthreads();
    if (tid == 0) {
      int acc = 0, bucket = 0;
      for (int bkt = 255; bkt >= 0; --bkt) {
        int c = (int)s_hist[bkt];
        if (acc + c >= need) { bucket = bkt; break; }
        acc += c;
      }
      s_misc[0] = prefix | ((uint)bucket << shift);
      s_misc[1] = (uint)(need - acc);
    }
    __syncthreads();
    prefix = s_misc[0];
    need = (int)s_misc[1];
    __syncthreads();
  }
  *need_out = need;  // how many values == T are part of the top `need_in`
  return prefix;
}

// ---------------------------------------------------------------------------
// Kernel 1: one workgroup per (b,c) plane. 256 threads = 8 wave32.
// Dynamic LDS: plane(64KB) + rmax(64KB) + hist(1KB) + misc.
// ---------------------------------------------------------------------------
__global__ void __launch_bounds__(256)
cn_plane_topk_kernel(const float* __restrict__ cls_logits,
                     float* __restrict__ cand_score,
                     int* __restrict__ cand_idx) {
  extern __shared__ unsigned char smem_raw[];
  float* s_plane = (float*)smem_raw;                 // 16384 f32
  float* s_rmax  = s_plane + CN_HW;                  // 16384 f32
  uint*  s_hist  = (uint*)(s_rmax + CN_HW);          // 256
  uint*  s_misc  = s_hist + 256;                     // 8

  const int tid = threadIdx.x;
  const int bc  = blockIdx.x;                        // b*C + c
  const float* __restrict__ src = cls_logits + (size_t)bc * CN_HW;

#if CN_USE_TDM
  // ---- Tensor Data Mover: DMA the whole 64KB plane into LDS -------------
  if (tid < 32) {  // single wave issues the TDM op (EXEC ignored by TDM)
    ull gaddr = (ull)(const void*)src;
    uint lds_byte = (uint)(size_t)(void*)s_plane;    // low 32b of LDS aperture
    u32x4 g0;
    g0[0] = 1u;                                      // count=1 valid descriptor
    g0[1] = lds_byte;                                // lds_addr
    g0[2] = (uint)(gaddr & 0xFFFFFFFFull);           // global_addr[31:0]
    g0[3] = (uint)((gaddr >> 32) & 0x01FFFFFFull) | (2u << 30);  // type=2
    i32x8 g1;
    g1[0] = 0x00020000;                              // data_size=4B, no mask
    g1[1] = (CN_W << 16);                            // tensor_dim0 lo16 = 128
    g1[2] = (CN_H << 16);                            // dim0 hi=0, dim1 lo16=128
    g1[3] = (CN_W << 16);                            // dim1 hi=0, tile_dim0=128
    g1[4] = CN_H;                                    // tile_dim1=128, tile_dim2=0
    g1[5] = CN_W;                                    // tensor_dim0_stride=128
    g1[6] = 0;                                       // stride hi, dim1_stride lo
    g1[7] = 0;
    i32x4 gz4 = {0, 0, 0, 0};                        // groups 2/3 unused (2-D)
    i32x8 gz8 = {0, 0, 0, 0, 0, 0, 0, 0};
    // clang-23 / amdgpu-toolchain form: 6 args (g0, g1, g2, g3, g_extra, cpol)
    __builtin_amdgcn_tensor_load_to_lds(g0, g1, gz4, gz4, gz8, 0);
    __builtin_amdgcn_s_wait_tensorcnt(0);
  }
  __syncthreads();
#else
  {  // fallback: vectorized manual staging
    const float4* s4 = (const float4*)src;
    float4* d4 = (float4*)s_plane;
    for (int i = tid; i < CN_HW / 4; i += 256) d4[i] = s4[i];
  }
  __syncthreads();
#endif

  // ---- 5x5 maxpool (separable) on raw logits: row pass ------------------
  for (int i = tid; i < CN_HW; i += 256) {
    const int r = i >> 7, c = i & 127;
    const int c0 = (c >= 2) ? c - 2 : 0;
    const int c1 = (c <= CN_W - 3) ? c + 2 : CN_W - 1;
    const float* row = s_plane + r * CN_W;
    float m = row[c0];
    for (int cc = c0 + 1; cc <= c1; ++cc) m = fmaxf(m, row[cc]);
    s_rmax[i] = m;
  }
  __syncthreads();

  // ---- column pass + keep mask + sigmoid (in place) ---------------------
  for (int i = tid; i < CN_HW; i += 256) {
    const int r = i >> 7, c = i & 127;
    const int r0 = (r >= 2) ? r - 2 : 0;
    const int r1 = (r <= CN_H - 3) ? r + 2 : CN_H - 1;
    float h = s_rmax[r0 * CN_W + c];
    for (int rr = r0 + 1; rr <= r1; ++rr) h = fmaxf(h, s_rmax[rr * CN_W + c]);
    const float v = s_plane[i];
    // maxpool(sigmoid(x)) == sigmoid(maxpool(x)): compare on logits.
    s_plane[i] = (h == v) ? cn_sigmoid(v) : 0.0f;
  }
  __syncthreads();

  // ---- exact per-class top-100 via radix select -------------------------
  int need_eq;
  const uint T = cn_radix_select(s_plane, CN_HW, CN_K, &need_eq,
                                 s_hist, s_misc, tid, 256);
  if (tid == 0) { s_misc[2] = 0u; s_misc[3] = 0u; }
  __syncthreads();

  float* __restrict__ out_s = cand_score + (size_t)bc * CN_K;
  int*   __restrict__ out_i = cand_idx + (size_t)bc * CN_K;

  for (int i = tid; i < CN_HW; i += 256) {
    const uint bts = __float_as_uint(s_plane[i]);
    if (bts > T) {
      const uint pos = atomicAdd(&s_misc[2], 1u);   // count(>T) < 100 by constr.
      out_s[pos] = s_plane[i];
      out_i[pos] = i;
    }
  }
  __syncthreads();
  const uint ngt = s_misc[2];
  for (int i = tid; i < CN_HW; i += 256) {
    const uint bts = __float_as_uint(s_plane[i]);
    if (bts == T) {
      const uint pos = ngt + atomicAdd(&s_misc[3], 1u);
      if (pos < CN_K) { out_s[pos] = s_plane[i]; out_i[pos] = i; }
    }
  }
}

// ---------------------------------------------------------------------------
// Kernel 2: one workgroup per batch. 8000 -> top-100 sorted + box decode.
// ---------------------------------------------------------------------------
__global__ void __launch_bounds__(256)
cn_finalize_kernel(const float* __restrict__ cand_score,
                   const int* __restrict__ cand_idx,
                   const float* __restrict__ txty,
                   const float* __restrict__ twth,
                   float* __restrict__ out) {
  __shared__ float s_sc[CN_CAND];        // 32000 B
  __shared__ uint  s_hist[256];
  __shared__ uint  s_misc[8];
  __shared__ ull   s_key[128];

  const int tid = threadIdx.x;
  const int b   = blockIdx.x;
  const float* __restrict__ csc = cand_score + (size_t)b * CN_CAND;
  const int*   __restrict__ cix = cand_idx + (size_t)b * CN_CAND;

  for (int j = tid; j < CN_CAND; j += 256) s_sc[j] = csc[j];
  __syncthreads();

  int need_eq;
  const uint T = cn_radix_select(s_sc, CN_CAND, CN_K, &need_eq,
                                 s_hist, s_misc, tid, 256);
  if (tid < 128) s_key[tid] = 0ull;
  if (tid == 0) { s_misc[2] = 0u; s_misc[3] = 0u; }
  __syncthreads();

  for (int j = tid; j < CN_CAND; j += 256) {
    const uint bts = __float_as_uint(s_sc[j]);
    if (bts > T) {
      const uint pos = atomicAdd(&s_misc[2], 1u);
      s_key[pos] = ((ull)bts << 32) | (uint)j;
    }
  }
  __syncthreads();
  const uint ngt = s_misc[2];
  for (int j = tid; j < CN_CAND; j += 256) {
    const uint bts = __float_as_uint(s_sc[j]);
    if (bts == T) {
      const uint pos = ngt + atomicAdd(&s_misc[3], 1u);
      if (pos < CN_K) s_key[pos] = ((ull)bts << 32) | (uint)j;
    }
  }

  // ---- bitonic sort of 128 keys, descending -----------------------------
  for (int size = 2; size <= 128; size <<= 1) {
    for (int stride = size >> 1; stride > 0; stride >>= 1) {
      __syncthreads();
      if (tid < 128) {
        const int p = tid ^ stride;
        if (p > tid) {
          const bool desc = ((tid & size) == 0);
          const ull a = s_key[tid], c = s_key[p];
          if (desc ? (a < c) : (a > c)) { s_key[tid] = c; s_key[p] = a; }
        }
      }
    }
  }
  __syncthreads();

  // ---- unpack winners, decode boxes, write flat outputs -----------------
  if (tid < CN_K) {
    const ull key = s_key[tid];
    const float score = __uint_as_float((uint)(key >> 32));
    const int j = (int)(key & 0xFFFFFFFFull);
    const int cls = j / CN_K;
    const int sp  = cix[j];
    const int x = sp & (CN_W - 1);
    const int y = sp >> 7;

    const size_t base = (size_t)b * 2 * CN_HW;
    const float tx = txty[base + sp];
    const float ty = txty[base + CN_HW + sp];
    const float tw = twth[base + sp];
    const float th = twth[base + CN_HW + sp];

    const float cx = (cn_sigmoid(tx) + (float)x) * CN_STRIDE;
    const float cy = (cn_sigmoid(ty) + (float)y) * CN_STRIDE;
    const float w  = __expf(tw) * CN_STRIDE;
    const float h  = __expf(th) * CN_STRIDE;

    const float x1 = fminf(fmaxf((cx - 0.5f * w) * CN_INV_INPUT, 0.0f), 1.0f);
    const float y1 = fminf(fmaxf((cy - 0.5f * h) * CN_INV_INPUT, 0.0f), 1.0f);
    const float x2 = fminf(fmaxf((cx + 0.5f * w) * CN_INV_INPUT, 0.0f), 1.0f);
    const float y2 = fminf(fmaxf((cy + 0.5f * h) * CN_INV_INPUT, 0.0f), 1.0f);

    const int o = b * CN_K + tid;
    out[o] = score;                                   // score   [B,K]
    float* bb = out + CN_B * CN_K + o * 4;            // bbox    [B,K,4]
    bb[0] = x1; bb[1] = y1; bb[2] = x2; bb[3] = y2;
    out[CN_B * CN_K * 5 + o] = (float)sp;             // inds    [B,K]
    out[CN_B * CN_K * 6 + o] = (float)cls;            // clses   [B,K]
  }
}

// ---------------------------------------------------------------------------
extern "C" void kernel_launch(void* const* d_in, const int* in_sizes, int n_in,
                              void* d_out, int out_size, void* d_ws, size_t ws_size,
                              hipStream_t stream) {
  const float* cls_logits = (const float*)d_in[0];  // [8,80,128,128]
  const float* txty       = (const float*)d_in[1];  // [8,2,128,128]
  const float* twth       = (const float*)d_in[2];  // [8,2,128,128]
  float* out = (float*)d_out;                       // 5600 floats

  float* cand_score = (float*)d_ws;                                    // 640*100
  int*   cand_idx   = (int*)((char*)d_ws + (size_t)CN_B * CN_C * CN_K * sizeof(float));

  const size_t smem1 = (size_t)CN_HW * sizeof(float) * 2   // plane + rmax
                     + 256 * sizeof(uint) + 8 * sizeof(uint);

  cn_plane_topk_kernel<<<CN_B * CN_C, 256, smem1, stream>>>(
      cls_logits, cand_score, cand_idx);
  cn_finalize_kernel<<<CN_B, 256, 0, stream>>>(
      cand_score, cand_idx, txty, twth, out);
}